// TensorAggregateLayer_51144470561241
// MI455X (gfx1250) — compile-verified
//
#include <hip/hip_runtime.h>
#include <hip/hip_bf16.h>

typedef __attribute__((ext_vector_type(2))) float v2f;
typedef __attribute__((ext_vector_type(8))) float v8f;

#define BB 4
#define NN 512
#define JJ 32
#define CC 64
#define NMAX 8
#define RCUT 5.0f
#define PI_F 3.14159265358979f
#define NPB 4   // nodes per block

__global__ __launch_bounds__(256) void TensorAggregateLayer_51144470561241_kernel(
    const float* __restrict__ x0, const float* __restrict__ x1, const float* __restrict__ x2,
    const float* __restrict__ rij, const float* __restrict__ W0, const float* __restrict__ W1,
    const float* __restrict__ W2, const int* __restrict__ neighbor,
    const unsigned char* __restrict__ mask, float* __restrict__ out)
{
    __shared__ float sW[3][NMAX * CC];          // 6 KB
    __shared__ float sRbf[NPB][JJ * NMAX];      // 4 KB  (rbf * fc * valid)
    __shared__ float sRij[NPB][JJ * 3];         // 1.5 KB
    __shared__ int   sNb[NPB][JJ];              // 0.5 KB
    __shared__ float sFn[3][NPB][JJ * CC];      // 96 KB (rbf @ W_r)

    const int tid = threadIdx.x;
    const int bn0 = blockIdx.x * NPB;           // 4 consecutive (b*N+n); N%4==0 so same batch

    // ---------------- Phase 1: W -> LDS, per-edge radial basis ----------------
    for (int i = tid; i < 3 * NMAX * CC; i += 256) {
        int r = i >> 9;                // / 512
        int rem = i & 511;
        const float* Wp = (r == 0) ? W0 : ((r == 1) ? W1 : W2);
        sW[r][rem] = Wp[rem];
    }
    if (tid < NPB * JJ) {
        int nl = tid >> 5, j = tid & 31;
        int bn = bn0 + nl;
        long ebase = (long)bn * JJ + j;
        float vx = rij[ebase * 3 + 0];
        float vy = rij[ebase * 3 + 1];
        float vz = rij[ebase * 3 + 2];
        sRij[nl][j * 3 + 0] = vx;
        sRij[nl][j * 3 + 1] = vy;
        sRij[nl][j * 3 + 2] = vz;
        sNb[nl][j] = neighbor[ebase];
        float d = __builtin_sqrtf(vx * vx + vy * vy + vz * vz);
        float validf = mask[ebase] ? 0.0f : 1.0f;
        float fc = (d < RCUT) ? 0.5f * (__cosf(PI_F * d * (1.0f / RCUT)) + 1.0f) : 0.0f;
        float pref = 0.6324555320336759f / fmaxf(d, 1e-9f) * fc * validf; // sqrt(2/RC)/d * fc * valid
        #pragma unroll
        for (int k = 0; k < NMAX; ++k)
            sRbf[nl][j * NMAX + k] = pref * __sinf((float)(k + 1) * (PI_F / RCUT) * d);
    }
    __syncthreads();

    // ---------------- Phase 2: fn_r = rbf(32x8) @ W_r(8x64) via WMMA f32 16x16x4 ----------------
    {
        const int w = tid >> 5, l = tid & 31;
        const int mt = w >> 2, nt = w & 3;       // 2 x 4 tile grid of the 32x64 result
        const int lhi = l >> 4, llo = l & 15;
        const int row = 16 * mt + llo;           // A-matrix M (j index)
        const int col = 16 * nt + llo;           // B/D N (channel)
        for (int nl = 0; nl < NPB; ++nl) {
            #pragma unroll
            for (int r = 0; r < 3; ++r) {
                v8f acc = {0.f, 0.f, 0.f, 0.f, 0.f, 0.f, 0.f, 0.f};
                #pragma unroll
                for (int kt = 0; kt < 2; ++kt) {
                    int kb = 4 * kt + 2 * lhi;   // f32 A/B layout: lanes 16-31 hold K+2
                    v2f a, b;
                    a.x = sRbf[nl][row * NMAX + kb];
                    a.y = sRbf[nl][row * NMAX + kb + 1];
                    b.x = sW[r][kb * CC + col];
                    b.y = sW[r][(kb + 1) * CC + col];
                    acc = __builtin_amdgcn_wmma_f32_16x16x4_f32(
                        false, a, false, b, (short)0, acc, false, false);
                }
                #pragma unroll
                for (int i = 0; i < 8; ++i) {    // D layout: VGPR i -> M = i + 8*(lane/16)
                    int jj = 16 * mt + i + 8 * lhi;
                    sFn[r][nl][jj * CC + col] = acc[i];
                }
            }
        }
    }
    __syncthreads();

    // ---------------- Phase 3: gather + tensor contractions, sum over neighbors ----------------
    {
        const int nl = tid >> 6, c = tid & 63;
        const int bn = bn0 + nl;
        const long rowb_batch = (long)(bn & ~(NN - 1)); // b*N

        float o0 = 0.f;
        float o1[3] = {0.f, 0.f, 0.f};
        float o2[9] = {0.f, 0.f, 0.f, 0.f, 0.f, 0.f, 0.f, 0.f, 0.f};

        for (int j = 0; j < JJ; ++j) {
            int nb = sNb[nl][j];
            long gi = (rowb_batch + nb) * CC + c;
            float g0v = x0[gi];
            float g1v[3], g2v[9], vv[3];
            #pragma unroll
            for (int p = 0; p < 3; ++p) g1v[p] = x1[gi * 3 + p];
            #pragma unroll
            for (int p = 0; p < 9; ++p) g2v[p] = x2[gi * 9 + p];
            #pragma unroll
            for (int p = 0; p < 3; ++p) vv[p] = sRij[nl][j * 3 + p];

            float fn0 = sFn[0][nl][j * CC + c] * g0v;  // = reference fn_r (valid & fc folded in rbf)
            float fn1 = sFn[1][nl][j * CC + c] * g0v;
            float fn2 = sFn[2][nl][j * CC + c] * g0v;

            float d1 = g1v[0] * vv[0] + g1v[1] * vv[1] + g1v[2] * vv[2];   // g1 . v
            float m2v[3];
            #pragma unroll
            for (int p = 0; p < 3; ++p)
                m2v[p] = g2v[p * 3 + 0] * vv[0] + g2v[p * 3 + 1] * vv[1] + g2v[p * 3 + 2] * vv[2];
            float vdm = vv[0] * m2v[0] + vv[1] * m2v[1] + vv[2] * m2v[2];  // v . g2 . v

            // out0: (0,0) + (1,1) + (2,2)
            o0 += fn0 * g0v + fn1 * d1 + fn2 * vdm;

            // out1: (0,1) + (1,0) + (1,2) + (2,1)
            float s1 = fn1 * g0v + fn2 * d1;
            #pragma unroll
            for (int p = 0; p < 3; ++p)
                o1[p] += s1 * vv[p] + fn0 * g1v[p] + fn1 * m2v[p];

            // out2: (0,2) + (1,1) + (2,0) + (2,2)
            float dp[3], cp[3];
            #pragma unroll
            for (int p = 0; p < 3; ++p) {
                dp[p] = fn2 * (g0v * vv[p] + m2v[p]);   // (0,2) + (2,2)
                cp[p] = fn1 * g1v[p];                   // (1,1)
            }
            #pragma unroll
            for (int p = 0; p < 3; ++p)
                #pragma unroll
                for (int q = 0; q < 3; ++q)
                    o2[p * 3 + q] += (dp[p] + cp[p]) * vv[q] + fn0 * g2v[p * 3 + q];
        }

        // outputs concatenated: out0 (B*N*C), out1 (B*N*C*3), out2 (B*N*C*9)
        size_t ob = (size_t)bn * CC + c;
        out[ob] = o0;
        float* out1p = out + (size_t)BB * NN * CC;
        float* out2p = out + (size_t)BB * NN * CC * 4;
        #pragma unroll
        for (int p = 0; p < 3; ++p) out1p[ob * 3 + p] = o1[p];
        #pragma unroll
        for (int p = 0; p < 9; ++p) out2p[ob * 9 + p] = o2[p];
    }
}

extern "C" void kernel_launch(void* const* d_in, const int* in_sizes, int n_in,
                              void* d_out, int out_size, void* d_ws, size_t ws_size,
                              hipStream_t stream) {
    const float* x0 = (const float*)d_in[0];
    const float* x1 = (const float*)d_in[1];
    const float* x2 = (const float*)d_in[2];
    const float* rij = (const float*)d_in[3];
    const float* W0 = (const float*)d_in[4];
    const float* W1 = (const float*)d_in[5];
    const float* W2 = (const float*)d_in[6];
    const int* neighbor = (const int*)d_in[7];
    const unsigned char* mask = (const unsigned char*)d_in[8];
    float* out = (float*)d_out;

    dim3 grid((BB * NN) / NPB);   // 512 blocks
    dim3 block(256);              // 8 wave32 waves
    TensorAggregateLayer_51144470561241_kernel<<<grid, block, 0, stream>>>(
        x0, x1, x2, rij, W0, W1, W2, neighbor, mask, out);
}